// TimeAttention_6339371729048
// MI455X (gfx1250) — compile-verified
//
#include <hip/hip_runtime.h>
#include <hip/hip_bf16.h>

// Problem constants (match reference)
#define DIM   128
#define TLEN  1024
#define NSEQ  32                       // B*F = 2*16
#define NROWS (NSEQ * TLEN)            // 32768
#define SEQSZ ((size_t)NROWS * DIM)    // floats per (rows x DIM) buffer
#define WINSZ 128
#define DFF   512

typedef float v2f __attribute__((ext_vector_type(2)));
typedef float v8f __attribute__((ext_vector_type(8)));

#define WMMA4(a, b, c) \
  __builtin_amdgcn_wmma_f32_16x16x4_f32(false, (a), false, (b), (short)0, (c), false, false)

__device__ __forceinline__ float wave_sum(float v) {
#pragma unroll
  for (int o = 16; o > 0; o >>= 1) v += __shfl_xor(v, o, 32);
  return v;
}

// A/B fragment loader for V_WMMA_F32_16X16X4_F32.
// Source: 16 vectors, row-major along K (ld = row stride).
// Layout per ISA: lanes 0-15 -> K = k0,k0+1 ; lanes 16-31 -> K = k0+2,k0+3.
__device__ __forceinline__ v2f frag_ld(const float* base, int ld, int lane, int k0) {
  int r  = lane & 15;
  int ks = k0 + ((lane >> 4) << 1);
  const float* p = base + (size_t)r * ld + ks;
  v2f f;
  f.x = p[0];
  f.y = p[1];
  return f;
}

// Transposed variant: element (k, n) lives at base[k*ld + n] (used for P @ V).
__device__ __forceinline__ v2f frag_ld_t(const float* base, int ld, int lane, int k0) {
  int r  = lane & 15;
  int ks = k0 + ((lane >> 4) << 1);
  v2f f;
  f.x = base[(size_t)ks * ld + r];
  f.y = base[(size_t)(ks + 1) * ld + r];
  return f;
}

// -------------------------------------------------------------------------
// K1: lift (per-channel affine), LayerNorm1, RoPE.  One wave per (n,t) row.
// Writes H (raw lift), V (=h1), Q (=rope(h1), also used as K).
// -------------------------------------------------------------------------
__global__ void k_lift_ln_rope(const float* __restrict__ x,
                               const float* __restrict__ w_lift,
                               const float* __restrict__ b_lift,
                               const float* __restrict__ g1,
                               const float* __restrict__ beta1,
                               float* __restrict__ H, float* __restrict__ Q,
                               float* __restrict__ V) {
  int wave = threadIdx.x >> 5, lane = threadIdx.x & 31;
  int row = blockIdx.x * 8 + wave;   // n*T + t
  int n = row >> 10, t = row & 1023;
  int b = n >> 4, f = n & 15;

  float xv = x[((size_t)b * TLEN + t) * 16 + f];
  int d0 = lane * 4;

  float4 wl = *(const float4*)(w_lift + f * DIM + d0);
  float4 bl = *(const float4*)(b_lift + f * DIM + d0);
  float h0 = fmaf(xv, wl.x, bl.x);
  float h1 = fmaf(xv, wl.y, bl.y);
  float h2 = fmaf(xv, wl.z, bl.z);
  float h3 = fmaf(xv, wl.w, bl.w);

  float mean = wave_sum(h0 + h1 + h2 + h3) * (1.0f / DIM);
  float c0 = h0 - mean, c1 = h1 - mean, c2 = h2 - mean, c3 = h3 - mean;
  float var = wave_sum(c0 * c0 + c1 * c1 + c2 * c2 + c3 * c3) * (1.0f / DIM);
  float rstd = rsqrtf(var + 1e-5f);

  float4 g  = *(const float4*)(g1 + d0);
  float4 be = *(const float4*)(beta1 + d0);
  float e0 = fmaf(c0 * rstd, g.x, be.x);
  float e1 = fmaf(c1 * rstd, g.y, be.y);
  float e2 = fmaf(c2 * rstd, g.z, be.z);
  float e3 = fmaf(c3 * rstd, g.w, be.w);

  size_t ro = (size_t)row * DIM + d0;
  *(float4*)(H + ro) = make_float4(h0, h1, h2, h3);
  *(float4*)(V + ro) = make_float4(e0, e1, e2, e3);

  // RoPE: pairs (d,d+1) within each 32-wide head; freq index i = (d%32)/2,
  // inv_freq[i] = 10000^(-i/16).  d0 is a multiple of 4 -> two whole pairs.
  int dd = d0 & 31;
  int i0 = dd >> 1;
  float fr0 = __powf(10000.0f, -(float)i0 * (1.0f / 16.0f));
  float fr1 = __powf(10000.0f, -(float)(i0 + 1) * (1.0f / 16.0f));
  float a0 = (float)t * fr0, a1 = (float)t * fr1;
  float s0 = sinf(a0), cc0 = cosf(a0);
  float s1 = sinf(a1), cc1 = cosf(a1);
  float q0 = e0 * cc0 - e1 * s0;
  float q1 = e0 * s0 + e1 * cc0;
  float q2 = e2 * cc1 - e3 * s1;
  float q3 = e2 * s1 + e3 * cc1;
  *(float4*)(Q + ro) = make_float4(q0, q1, q2, q3);
}

// -------------------------------------------------------------------------
// K2: fused QKV projection, register-blocked: one wave = 16x64 output strip
// (4 N-tiles), so each A fragment is loaded once and reused 4x.
// Out cols 0..383 = [qp | kp | vp]; A = Q for qp & kp (k == q), A = V for vp.
// -------------------------------------------------------------------------
__global__ void k_qkv(const float* __restrict__ Q, const float* __restrict__ V,
                      const float* __restrict__ w_in, const float* __restrict__ b_in,
                      float* __restrict__ QP, float* __restrict__ KP,
                      float* __restrict__ VP) {
  int wave = threadIdx.x >> 5, lane = threadIdx.x & 31;
  int gw = blockIdx.x * 8 + wave;     // < 2048*6
  int mtile = gw / 6;
  int ot4   = gw % 6;                 // block of 4 consecutive 16-col tiles

  const float* A  = ((ot4 < 4) ? Q : V) + (size_t)mtile * 16 * DIM;
  const float* W0 = w_in + (size_t)ot4 * 4 * 16 * DIM;

  v8f acc[4] = {};
#pragma unroll 4
  for (int k0 = 0; k0 < DIM; k0 += 4) {
    v2f a = frag_ld(A, DIM, lane, k0);
#pragma unroll
    for (int u = 0; u < 4; ++u) {
      v2f w = frag_ld(W0 + (size_t)u * 16 * DIM, DIM, lane, k0);
      acc[u] = WMMA4(a, w, acc[u]);
    }
  }

  int ncol = lane & 15, half = lane >> 4;
  float* Out = (ot4 < 2) ? QP : ((ot4 < 4) ? KP : VP);
#pragma unroll
  for (int u = 0; u < 4; ++u) {
    int otile = ot4 * 4 + u;
    int o = otile * 16 + ncol;
    float bias = b_in[o];
    int oc = o & 127;
#pragma unroll
    for (int r = 0; r < 8; ++r)
      Out[((size_t)mtile * 16 + r + 8 * half) * DIM + oc] = acc[u][r] + bias;
  }
}

// -------------------------------------------------------------------------
// K3: windowed attention (0 <= i-j <= 128), flash-lite per wave.
// One wave handles (seq n, head h, 16-query tile qt).  Q tile fragments are
// hoisted into registers once; scores for the 9 candidate key tiles -> LDS
// (16x144), exact softmax, then P@V via WMMA.  y overwrites QP (disjoint
// region per wave).
// -------------------------------------------------------------------------
#define SCOLS 144
__global__ void k_attn(const float* __restrict__ QPin, const float* __restrict__ KP,
                       const float* __restrict__ VP, float* __restrict__ Y) {
  __shared__ float S[4][16 * SCOLS];
  int wave = threadIdx.x >> 5, lane = threadIdx.x & 31;
  int unit = blockIdx.x * 4 + wave;
  int n  = unit >> 8;          // 4 heads * 64 qtiles per sequence
  int h  = (unit >> 6) & 3;
  int qt = unit & 63;
  float* Sw = S[wave];
  int ncol = lane & 15, half = lane >> 4;
  const float scale = 0.17677669529663687f;  // 1/sqrt(32)

  const float* Qb = QPin + ((size_t)n * TLEN + qt * 16) * DIM + h * 32;

  // Hoist the 16x32 Q tile: 8 fragments (16 floats/lane), reused for all kt.
  v2f qa[8];
#pragma unroll
  for (int k0 = 0; k0 < 32; k0 += 4) qa[k0 >> 2] = frag_ld(Qb, DIM, lane, k0);

  // Pass 1: score tiles
  for (int r = 0; r < 9; ++r) {
    int kt = qt - 8 + r;
    if (kt < 0) {
#pragma unroll
      for (int rr = 0; rr < 8; ++rr)
        Sw[(rr + 8 * half) * SCOLS + r * 16 + ncol] = -1e30f;
      continue;
    }
    const float* Kb = KP + ((size_t)n * TLEN + kt * 16) * DIM + h * 32;
    v8f acc = {};
#pragma unroll
    for (int k0 = 0; k0 < 32; k0 += 4) {
      v2f k = frag_ld(Kb, DIM, lane, k0);
      acc = WMMA4(qa[k0 >> 2], k, acc);
    }
#pragma unroll
    for (int rr = 0; rr < 8; ++rr) {
      int i = qt * 16 + rr + 8 * half;
      int j = kt * 16 + ncol;
      bool ok = (i >= j) && ((i - j) <= WINSZ);
      Sw[(rr + 8 * half) * SCOLS + r * 16 + ncol] = ok ? acc[rr] * scale : -1e30f;
    }
  }
  __syncthreads();

  // Softmax: lane handles row (lane&15), half-row chunk of 72 columns.
  {
    float* row = Sw + (lane & 15) * SCOLS + half * 72;
    float mx = -1e30f;
    for (int c = 0; c < 72; ++c) mx = fmaxf(mx, row[c]);
    mx = fmaxf(mx, __shfl_xor(mx, 16, 32));
    float sum = 0.0f;
    for (int c = 0; c < 72; ++c) {
      float e = __expf(row[c] - mx);
      row[c] = e;
      sum += e;
    }
    sum += __shfl_xor(sum, 16, 32);
    float inv = 1.0f / sum;
    for (int c = 0; c < 72; ++c) row[c] *= inv;
  }
  __syncthreads();

  // Pass 2: y = P @ V  (K over valid key positions, N = 32 head dims)
  v8f acc0 = {}, acc1 = {};
  int r0 = (qt >= 8) ? 0 : (8 - qt);
  for (int r = r0; r < 9; ++r) {
    int kbase = (qt - 8 + r) * 16;
    const float* Vb = VP + ((size_t)n * TLEN + kbase) * DIM + h * 32;
#pragma unroll
    for (int k = 0; k < 16; k += 4) {
      v2f a  = frag_ld(Sw, SCOLS, lane, r * 16 + k);
      v2f b0 = frag_ld_t(Vb, DIM, lane, k);
      v2f b1 = frag_ld_t(Vb + 16, DIM, lane, k);
      acc0 = WMMA4(a, b0, acc0);
      acc1 = WMMA4(a, b1, acc1);
    }
  }
  float* Yb = Y + ((size_t)n * TLEN + qt * 16) * DIM + h * 32;
#pragma unroll
  for (int rr = 0; rr < 8; ++rr) {
    Yb[(rr + 8 * half) * DIM + ncol]      = acc0[rr];
    Yb[(rr + 8 * half) * DIM + 16 + ncol] = acc1[rr];
  }
}

// -------------------------------------------------------------------------
// K4: h2 = h + y @ w_out^T + b_out.  Register-blocked 16x64 per wave.
// -------------------------------------------------------------------------
__global__ void k_outproj(const float* __restrict__ Y, const float* __restrict__ H,
                          const float* __restrict__ w_out,
                          const float* __restrict__ b_out, float* __restrict__ H2) {
  int wave = threadIdx.x >> 5, lane = threadIdx.x & 31;
  int gw = blockIdx.x * 8 + wave;     // < 2048*2
  int mtile = gw >> 1;
  int ot4   = gw & 1;                 // 4 N-tiles per wave (cols 0..63 / 64..127)

  const float* A  = Y + (size_t)mtile * 16 * DIM;
  const float* W0 = w_out + (size_t)ot4 * 4 * 16 * DIM;

  v8f acc[4] = {};
#pragma unroll 4
  for (int k0 = 0; k0 < DIM; k0 += 4) {
    v2f a = frag_ld(A, DIM, lane, k0);
#pragma unroll
    for (int u = 0; u < 4; ++u) {
      v2f w = frag_ld(W0 + (size_t)u * 16 * DIM, DIM, lane, k0);
      acc[u] = WMMA4(a, w, acc[u]);
    }
  }

  int ncol = lane & 15, half = lane >> 4;
#pragma unroll
  for (int u = 0; u < 4; ++u) {
    int o = (ot4 * 4 + u) * 16 + ncol;
    float bias = b_out[o];
#pragma unroll
    for (int r = 0; r < 8; ++r) {
      size_t idx = ((size_t)mtile * 16 + r + 8 * half) * DIM + o;
      H2[idx] = H[idx] + acc[u][r] + bias;
    }
  }
}

// -------------------------------------------------------------------------
// K5: fused LN2 + FFN (gelu exact) + residual + score head + final output.
// One block (8 waves) per 16-row tile.  FFN1 is register-blocked: the LDS
// h3 fragment is loaded once per K-step and reused for 4 weight tiles.
// -------------------------------------------------------------------------
__global__ void k_ffn(const float* __restrict__ H2, const float* __restrict__ w_ffn1,
                      const float* __restrict__ b_ffn1,
                      const float* __restrict__ w_ffn2,
                      const float* __restrict__ b_ffn2, const float* __restrict__ w_s,
                      const float* __restrict__ b_s, const float* __restrict__ x,
                      float* __restrict__ out) {
  __shared__ float sh3[16 * DIM];    // h3 tile, later reused for h4
  __shared__ float smid[16 * DFF];   // gelu(FFN1) tile
  int wave = threadIdx.x >> 5, lane = threadIdx.x & 31;
  int mtile = blockIdx.x;
  size_t rowbase = (size_t)mtile * 16;

  // Phase A: LN2 (gamma=1, beta=0) for 2 rows per wave
#pragma unroll
  for (int rr = 0; rr < 2; ++rr) {
    int lr = wave * 2 + rr;
    const float* hr = H2 + (rowbase + lr) * DIM;
    int d0 = lane * 4;
    float4 hv = *(const float4*)(hr + d0);
    float mean = wave_sum(hv.x + hv.y + hv.z + hv.w) * (1.0f / DIM);
    float cx = hv.x - mean, cy = hv.y - mean, cz = hv.z - mean, cw = hv.w - mean;
    float var = wave_sum(cx * cx + cy * cy + cz * cz + cw * cw) * (1.0f / DIM);
    float rstd = rsqrtf(var + 1e-5f);
    sh3[lr * DIM + d0 + 0] = cx * rstd;
    sh3[lr * DIM + d0 + 1] = cy * rstd;
    sh3[lr * DIM + d0 + 2] = cz * rstd;
    sh3[lr * DIM + d0 + 3] = cw * rstd;
  }
  __syncthreads();

  int ncol = lane & 15, half = lane >> 4;

  // Phase B: mid = gelu(h3 @ w_ffn1^T + b1); wave owns 4 N-tiles of the 512.
  {
    const float* WB = w_ffn1 + (size_t)wave * 4 * 16 * DIM;
    v8f acc[4] = {};
#pragma unroll 4
    for (int k0 = 0; k0 < DIM; k0 += 4) {
      v2f a = frag_ld(sh3, DIM, lane, k0);
#pragma unroll
      for (int u = 0; u < 4; ++u) {
        v2f w = frag_ld(WB + (size_t)u * 16 * DIM, DIM, lane, k0);
        acc[u] = WMMA4(a, w, acc[u]);
      }
    }
#pragma unroll
    for (int u = 0; u < 4; ++u) {
      int nt = wave * 4 + u;
      float bias = b_ffn1[nt * 16 + ncol];
#pragma unroll
      for (int rr = 0; rr < 8; ++rr) {
        float vv = acc[u][rr] + bias;
        float g = 0.5f * vv * (1.0f + erff(vv * 0.70710678118654752f));
        smid[(rr + 8 * half) * DFF + nt * 16 + ncol] = g;
      }
    }
  }
  __syncthreads();

  // Phase C: h4 = h2 + mid @ w_ffn2^T + b2; wave handles output N-tile `wave`
  {
    const float* W = w_ffn2 + (size_t)wave * 16 * DFF;
    v8f acc = {};
#pragma unroll 4
    for (int k0 = 0; k0 < DFF; k0 += 4) {
      v2f a = frag_ld(smid, DFF, lane, k0);
      v2f w = frag_ld(W, DFF, lane, k0);
      acc = WMMA4(a, w, acc);
    }
    int o = wave * 16 + ncol;
    float bias = b_ffn2[o];
#pragma unroll
    for (int rr = 0; rr < 8; ++rr) {
      int lr = rr + 8 * half;
      float h4 = H2[(rowbase + lr) * DIM + o] + acc[rr] + bias;
      sh3[lr * DIM + o] = h4;   // reuse sh3 as h4 tile
    }
  }
  __syncthreads();

  // Phase D: s = h4 . w_s + b_s ; out = x + s (scatter back to (B,T,F))
#pragma unroll
  for (int rr = 0; rr < 2; ++rr) {
    int lr = wave * 2 + rr;
    int d0 = lane * 4;
    float p = sh3[lr * DIM + d0 + 0] * w_s[d0 + 0] +
              sh3[lr * DIM + d0 + 1] * w_s[d0 + 1] +
              sh3[lr * DIM + d0 + 2] * w_s[d0 + 2] +
              sh3[lr * DIM + d0 + 3] * w_s[d0 + 3];
    p = wave_sum(p);
    if (lane == 0) {
      size_t rowg = rowbase + lr;
      int n = (int)(rowg >> 10), t = (int)(rowg & 1023);
      int b = n >> 4, f = n & 15;
      size_t xi = ((size_t)b * TLEN + t) * 16 + f;
      out[xi] = x[xi] + p + b_s[0];
    }
  }
}

// -------------------------------------------------------------------------
extern "C" void kernel_launch(void* const* d_in, const int* in_sizes, int n_in,
                              void* d_out, int out_size, void* d_ws, size_t ws_size,
                              hipStream_t stream) {
  (void)in_sizes; (void)n_in; (void)out_size; (void)ws_size;
  const float* x      = (const float*)d_in[0];
  const float* w_lift = (const float*)d_in[1];
  const float* b_lift = (const float*)d_in[2];
  const float* g1     = (const float*)d_in[3];
  const float* beta1  = (const float*)d_in[4];
  const float* w_in   = (const float*)d_in[5];
  const float* b_in   = (const float*)d_in[6];
  const float* w_out  = (const float*)d_in[7];
  const float* b_out  = (const float*)d_in[8];
  const float* w_ffn1 = (const float*)d_in[9];
  const float* b_ffn1 = (const float*)d_in[10];
  const float* w_ffn2 = (const float*)d_in[11];
  const float* b_ffn2 = (const float*)d_in[12];
  const float* w_s    = (const float*)d_in[13];
  const float* b_s    = (const float*)d_in[14];
  float* out = (float*)d_out;

  float* ws = (float*)d_ws;   // 7 x 16 MiB fp32 buffers = 112 MiB
  float* H  = ws + 0 * SEQSZ;
  float* Q  = ws + 1 * SEQSZ;
  float* V  = ws + 2 * SEQSZ;
  float* QP = ws + 3 * SEQSZ;
  float* KP = ws + 4 * SEQSZ;
  float* VP = ws + 5 * SEQSZ;
  float* H2 = ws + 6 * SEQSZ;

  k_lift_ln_rope<<<NROWS / 8, 256, 0, stream>>>(x, w_lift, b_lift, g1, beta1, H, Q, V);
  k_qkv<<<(NROWS / 16) * 6 / 8, 256, 0, stream>>>(Q, V, w_in, b_in, QP, KP, VP);
  k_attn<<<(NSEQ * 4 * (TLEN / 16)) / 4, 128, 0, stream>>>(QP, KP, VP, QP);  // y -> QP
  k_outproj<<<(NROWS / 16) * 2 / 8, 256, 0, stream>>>(QP, H, w_out, b_out, H2);
  k_ffn<<<NROWS / 16, 256, 0, stream>>>(H2, w_ffn1, b_ffn1, w_ffn2, b_ffn2, w_s, b_s, x, out);
}